// MPOLinear_6528350290553
// MI455X (gfx1250) — compile-verified
//
#include <hip/hip_runtime.h>

// MPO linear (x:[2048,4096] f32, 3 MPO cores, bias) fully fused.
// All three einsum stages run as f32 WMMA (v_wmma_f32_16x16x4_f32); t1/t2
// intermediates live in LDS only. One workgroup (16 waves, 512 threads) owns
// a 16-token tile and an 8-wide m-half of the 4096 outputs (grid.y = 2), so
// per-lane accumulators (64+64 VGPRs) fit the 4-waves/SIMD ~256-VGPR budget
// without scratch spills.

typedef __attribute__((ext_vector_type(2))) float v2f;
typedef __attribute__((ext_vector_type(8))) float v8f;

#define DLEG      16
#define IN_SIZE   4096
#define OUT_SIZE  4096
#define BTILE     16
#define T1_STRIDE 290                  // floats per m-slice of t1 (16*18 + pad)
#define T2_STRIDE 290                  // floats per (mm,n) tile of t2
#define T1_FLOATS (16 * T1_STRIDE)     // 4640  (all 16 m kept; stage1 is cheap)
#define T2_FLOATS (128 * T2_STRIDE)    // 37120 (8 local m x 16 n tiles)
#define SMEM_BYTES ((size_t)(T1_FLOATS + T2_FLOATS) * sizeof(float)) // 167,040 B

__device__ __forceinline__ v8f wmma_f32(v2f a, v2f b, v8f c) {
  // 8 args: (neg_a, A, neg_b, B, c_mod, C, reuse_a, reuse_b)
  return __builtin_amdgcn_wmma_f32_16x16x4_f32(false, a, false, b, (short)0, c,
                                               false, false);
}

__global__ __launch_bounds__(512, 1)
void mpo_linear_f32_wmma(const float* __restrict__ x,
                         const float* __restrict__ fc,   // [i][r][m]    16^3
                         const float* __restrict__ mc,   // [j][r][s][n] 16^4
                         const float* __restrict__ lc,   // [k][s][o]    16^3
                         const float* __restrict__ bias, // [4096]
                         float* __restrict__ out,        // [B][4096]
                         int ntok)
{
  extern __shared__ float smem[];
  float* t1 = smem;               // t1[m*290 + b*18 + r]        (all 16 m)
  float* t2 = smem + T1_FLOATS;   // t2[(mm*16+n)*290 + b*18 + s] (8 local m)

  const int tid  = threadIdx.x;
  const int lane = tid & 31;
  const int wave = tid >> 5;          // 0..15
  const int half = lane >> 4;         // 0/1 (wave32 halves)
  const int l15  = lane & 15;
  const int K0   = half * 2;          // A/B K offset within each K=4 chunk
  const int drow = half * 8;          // C/D row base per half

  const int b0   = blockIdx.x * BTILE;
  const int m0   = blockIdx.y * 8;    // this block's m-half
  const int s_w  = wave;              // stage1: r-tile, stage2: s-tile
  const int mm3  = wave & 7;          // stage3: local m
  const int n3_0 = (wave >> 3) * 8;   // stage3: n base (8 tiles/wave)

  const v8f vzero = {};
  v8f acc2[8];                        // stage-2 accumulators (one k at a time)
  v8f accO[8];                        // output accumulators (across all k)
  #pragma unroll
  for (int i = 0; i < 8; ++i) accO[i] = vzero;

  for (int k = 0; k < DLEG; ++k) {
    #pragma unroll
    for (int i = 0; i < 8; ++i) acc2[i] = vzero;

    for (int j = 0; j < DLEG; ++j) {
      // ---- Stage 1: t1_kj[b,(r,m)] = sum_i x[b,i,j,k] * fc[i,(r,m)]
      //      wave computes the r = s_w column tile (M=b, N=m, K=i)
      v8f d1 = vzero;
      const float* xk  = x  + (size_t)(b0 + l15) * IN_SIZE + j * 16 + k;
      const float* fcb = fc + s_w * 16 + l15;
      #pragma unroll
      for (int c = 0; c < 4; ++c) {
        const int i0 = c * 4 + K0;
        v2f a, b;
        a.x = xk [(i0    ) * 256];
        a.y = xk [(i0 + 1) * 256];
        b.x = fcb[(i0    ) * 256];
        b.y = fcb[(i0 + 1) * 256];
        d1 = wmma_f32(a, b, d1);
      }
      // dump t1 tile: m = l15 (col), rows b = v+drow, r = s_w
      #pragma unroll
      for (int v = 0; v < 8; ++v)
        t1[l15 * T1_STRIDE + (v + drow) * 18 + s_w] = d1[v];
      __syncthreads();

      // ---- Stage 2 (accumulate over j): acc2[mm] += t1[b,r,m0+mm] * mc[j,r,(s,n)]
      //      wave handles s = s_w for the block's 8 local m; B reused across m.
      const float* mcb = mc + (size_t)j * 4096 + s_w * 16 + l15;
      v2f bo[4];
      #pragma unroll
      for (int c = 0; c < 4; ++c) {
        const int r0 = c * 4 + K0;
        bo[c].x = mcb[(r0    ) * 256];
        bo[c].y = mcb[(r0 + 1) * 256];
      }
      #pragma unroll
      for (int mm = 0; mm < 8; ++mm) {
        const float* t1m = t1 + (m0 + mm) * T1_STRIDE + l15 * 18;
        #pragma unroll
        for (int c = 0; c < 4; ++c) {
          const int r0 = c * 4 + K0;
          v2f a = *(const v2f*)(t1m + r0);      // 8B-aligned ds_load_b64
          acc2[mm] = wmma_f32(a, bo[c], acc2[mm]);
        }
      }
      __syncthreads();   // t1 consumed before next j overwrites it
    }

    // ---- dump t2_k tiles: t2[(mm,n)][b][s], this wave owns s = s_w
    #pragma unroll
    for (int mm = 0; mm < 8; ++mm) {
      #pragma unroll
      for (int v = 0; v < 8; ++v)
        t2[(mm * 16 + l15) * T2_STRIDE + (v + drow) * 18 + s_w] = acc2[mm][v];
    }
    __syncthreads();

    // ---- Stage 3: accO[nn] += t2_k[b, mm3, s, n] * lc[k,s,o]  (K = s)
    const float* lcb = lc + k * 256 + l15;
    v2f bl[4];
    #pragma unroll
    for (int c = 0; c < 4; ++c) {
      const int s0 = c * 4 + K0;
      bl[c].x = lcb[(s0    ) * 16];
      bl[c].y = lcb[(s0 + 1) * 16];
    }
    #pragma unroll
    for (int nn = 0; nn < 8; ++nn) {
      const float* t2t = t2 + (mm3 * 16 + (n3_0 + nn)) * T2_STRIDE + l15 * 18;
      #pragma unroll
      for (int c = 0; c < 4; ++c) {
        const int s0 = c * 4 + K0;
        v2f a = *(const v2f*)(t2t + s0);        // 8B-aligned ds_load_b64
        accO[nn] = wmma_f32(a, bl[c], accO[nn]);
      }
    }
    __syncthreads();   // t2 (and t1) consumed before next k
  }

  // ---- epilogue: out[b, (m0+mm3)*256 + n*16 + o] = accO + bias
  #pragma unroll
  for (int nn = 0; nn < 8; ++nn) {
    const int col = (m0 + mm3) * 256 + (n3_0 + nn) * 16 + l15;   // o = l15
    const float bv = bias[col];
    #pragma unroll
    for (int v = 0; v < 8; ++v) {
      const int b = b0 + v + drow;
      if (b < ntok) out[(size_t)b * OUT_SIZE + col] = accO[nn][v] + bv;
    }
  }
}

extern "C" void kernel_launch(void* const* d_in, const int* in_sizes, int n_in,
                              void* d_out, int out_size, void* d_ws, size_t ws_size,
                              hipStream_t stream) {
  const float* x    = (const float*)d_in[0];
  const float* fc   = (const float*)d_in[1];
  const float* mc   = (const float*)d_in[2];
  const float* lc   = (const float*)d_in[3];
  const float* bias = (const float*)d_in[4];
  float* out = (float*)d_out;

  const int ntok   = in_sizes[0] / IN_SIZE;            // 2048
  const int blocks = (ntok + BTILE - 1) / BTILE;       // 128

  // Dynamic LDS > 64KB needs the opt-in attribute (host-side, capture-safe).
  hipFuncSetAttribute(reinterpret_cast<const void*>(mpo_linear_f32_wmma),
                      hipFuncAttributeMaxDynamicSharedMemorySize,
                      (int)SMEM_BYTES);

  mpo_linear_f32_wmma<<<dim3(blocks, 2), dim3(512), SMEM_BYTES, stream>>>(
      x, fc, mc, lc, bias, out, ntok);
}